// _MultiHeadGraphAttentionLayer_89919435309473
// MI455X (gfx1250) — compile-verified
//
#include <hip/hip_runtime.h>
#include <hip/hip_bf16.h>

typedef float v2f __attribute__((ext_vector_type(2)));
typedef float v8f __attribute__((ext_vector_type(8)));

#define HEADS 4
#define DDIM  64
#define COLS  256   // H*D == IN == OUT == 256

// ---------------------------------------------------------------------------
// Repack the 256x256 weight view into K-pair-interleaved layout:
//   Bp[(k>>1)*512 + n*2 + (k&1)] = B[k*strideI + (n>>6)*strideH + (n&63)]
// so a WMMA B-fragment (two consecutive K for one column) is one b64 load.
// ---------------------------------------------------------------------------
__global__ void gat_pack_B(const float* __restrict__ B, float* __restrict__ Bp,
                           int strideI, int strideH)
{
    const int idx = blockIdx.x * blockDim.x + threadIdx.x;  // k*256 + n
    if (idx >= COLS * COLS) return;
    const int k = idx >> 8;
    const int n = idx & 255;
    const float v = B[(size_t)k * strideI + (n >> 6) * strideH + (n & 63)];
    Bp[(size_t)(k >> 1) * 512 + n * 2 + (k & 1)] = v;
}

// ---------------------------------------------------------------------------
// WMMA f32 GEMM: C[M x 256] = A[M x 256] * B(packed), one wave per 16x64 strip.
// A-fragment layout (ISA 7.12.2, 32-bit A 16x4): lanes 0-15 rows, VGPR0 holds
// K=k0 (lo half) / K=k0+2 (hi half), VGPR1 holds K=k0+1 / K=k0+3.
// B/C/D: rows striped across lanes within a VGPR.
// ---------------------------------------------------------------------------
__global__ void __launch_bounds__(32)
gat_gemm_wmma_f32(const float* __restrict__ A, const float* __restrict__ Bp,
                  float* __restrict__ C, int M)
{
    const int m0   = blockIdx.x * 16;
    const int n0   = blockIdx.y * 64;
    const int lane = threadIdx.x & 31;
    const int half = lane >> 4;       // 0 or 1
    const int l    = lane & 15;

    int arow = m0 + l;
    if (arow >= M) arow = M - 1;      // clamp (stores guarded below)
    const float* Arow = A + (size_t)arow * COLS;

    // per-lane B base: k-pair p = (k0>>1)+half, columns n0+16t+l
    // addr = Bp + p*512 + n*2  =  (Bp + half*512 + (n0+l)*2) + k0*256 + t*32
    const float* Bbase = Bp + half * 512 + (size_t)(n0 + l) * 2;

    v8f acc[4];
    acc[0] = (v8f){}; acc[1] = (v8f){}; acc[2] = (v8f){}; acc[3] = (v8f){};

    for (int k0 = 0; k0 < COLS; k0 += 4) {
        const v2f a = *(const v2f*)(Arow + k0 + 2 * half);
        const float* bk = Bbase + (size_t)k0 * 256;
#pragma unroll
        for (int t = 0; t < 4; ++t) {
            const v2f b = *(const v2f*)(bk + t * 32);
            acc[t] = __builtin_amdgcn_wmma_f32_16x16x4_f32(
                false, a, false, b, (short)0, acc[t], false, false);
        }
    }

    float* Crow = C + (size_t)(m0 + 8 * half) * COLS + n0 + l;
    if (m0 + 16 <= M) {
        // full tile: unguarded stores
#pragma unroll
        for (int v = 0; v < 8; ++v) {
            float* cp = Crow + (size_t)v * COLS;
#pragma unroll
            for (int t = 0; t < 4; ++t) cp[t * 16] = acc[t][v];
        }
    } else {
#pragma unroll
        for (int v = 0; v < 8; ++v) {
            if (m0 + 8 * half + v < M) {
                float* cp = Crow + (size_t)v * COLS;
#pragma unroll
                for (int t = 0; t < 4; ++t) cp[t * 16] = acc[t][v];
            }
        }
    }
}

// ---------------------------------------------------------------------------
// Per-node attention scores: s_src[n,h] = h[n,h,:].a_src[h], s_dst likewise.
// ---------------------------------------------------------------------------
__global__ void gat_scores(const float* __restrict__ h,
                           const float* __restrict__ attn,
                           float* __restrict__ s_src, float* __restrict__ s_dst,
                           int Nn)
{
    const int idx = blockIdx.x * blockDim.x + threadIdx.x;  // n*H + hh
    if (idx >= Nn * HEADS) return;
    const int n  = idx >> 2;
    const int hh = idx & 3;
    const float* hp = h + (size_t)n * COLS + hh * DDIM;
    const float* ap = attn + hh * (2 * DDIM);
    float ss = 0.f, sd = 0.f;
#pragma unroll 8
    for (int d = 0; d < DDIM; ++d) {
        const float v = hp[d];
        ss += v * ap[d];
        sd += v * ap[DDIM + d];
    }
    s_src[idx] = ss;
    s_dst[idx] = sd;
}

__device__ __forceinline__ void atomicMaxFloat(float* addr, float val)
{
    if (val >= 0.f)
        atomicMax((int*)addr, __float_as_int(val));
    else
        atomicMin((unsigned int*)addr, __float_as_uint(val));
}

// ---------------------------------------------------------------------------
// init: out_h = 0, alpha_sum = 0, e_max = -FLT_MAX
// ---------------------------------------------------------------------------
__global__ void gat_init(float* __restrict__ out_h, float* __restrict__ alpha_sum,
                         float* __restrict__ e_max, int Nn)
{
    const int i = blockIdx.x * blockDim.x + threadIdx.x;
    if (i < Nn * COLS) out_h[i] = 0.f;
    if (i < Nn * HEADS) {
        alpha_sum[i] = 0.f;
        e_max[i]     = -3.402823466e38f;
    }
}

// ---------------------------------------------------------------------------
// One thread per edge, all 4 heads vectorized:
// e[ei,:] = leakyrelu(s_src[src,:]+s_dst[dst,:]); segment-max into e_max[dst,:]
// ---------------------------------------------------------------------------
__global__ void gat_edge_score(const int* __restrict__ src, const int* __restrict__ dst,
                               const float* __restrict__ s_src, const float* __restrict__ s_dst,
                               float* __restrict__ e, float* __restrict__ e_max, int Ecnt)
{
    const int ei = blockIdx.x * blockDim.x + threadIdx.x;
    if (ei >= Ecnt) return;
    const int s = src[ei];
    const int d = dst[ei];
    const float4 ss = *(const float4*)(s_src + (size_t)s * HEADS);
    const float4 sd = *(const float4*)(s_dst + (size_t)d * HEADS);
    float4 v;
    v.x = ss.x + sd.x; v.y = ss.y + sd.y; v.z = ss.z + sd.z; v.w = ss.w + sd.w;
    v.x = (v.x > 0.f) ? v.x : 0.2f * v.x;
    v.y = (v.y > 0.f) ? v.y : 0.2f * v.y;
    v.z = (v.z > 0.f) ? v.z : 0.2f * v.z;
    v.w = (v.w > 0.f) ? v.w : 0.2f * v.w;
    *(float4*)(e + (size_t)ei * HEADS) = v;
    float* em = e_max + (size_t)d * HEADS;
    atomicMaxFloat(em + 0, v.x);
    atomicMaxFloat(em + 1, v.y);
    atomicMaxFloat(em + 2, v.z);
    atomicMaxFloat(em + 3, v.w);
}

// ---------------------------------------------------------------------------
// alpha = exp(min(e - e_max[dst], 20)); segment-sum into alpha_sum[dst,:]
// (alpha overwrites e in place)
// ---------------------------------------------------------------------------
__global__ void gat_edge_alpha(const int* __restrict__ dst,
                               float* __restrict__ e, const float* __restrict__ e_max,
                               float* __restrict__ alpha_sum, int Ecnt)
{
    const int ei = blockIdx.x * blockDim.x + threadIdx.x;
    if (ei >= Ecnt) return;
    const int d = dst[ei];
    float4 ev       = *(const float4*)(e + (size_t)ei * HEADS);
    const float4 em = *(const float4*)(e_max + (size_t)d * HEADS);
    ev.x = expf(fminf(ev.x - em.x, 20.0f));
    ev.y = expf(fminf(ev.y - em.y, 20.0f));
    ev.z = expf(fminf(ev.z - em.z, 20.0f));
    ev.w = expf(fminf(ev.w - em.w, 20.0f));
    *(float4*)(e + (size_t)ei * HEADS) = ev;
    float* as = alpha_sum + (size_t)d * HEADS;
    atomicAdd(as + 0, ev.x);
    atomicAdd(as + 1, ev.y);
    atomicAdd(as + 2, ev.z);
    atomicAdd(as + 3, ev.w);
}

// ---------------------------------------------------------------------------
// out_h[dst, c] += (alpha / (alpha_sum[dst]+1e-8)) * h[src, c]
// 64 lanes x float4 per edge; 4 edges per 256-thread block.
// Gathers and atomics are L2-resident (h, out_h ~51MB each, L2 = 192MB).
// ---------------------------------------------------------------------------
__global__ void __launch_bounds__(256)
gat_aggregate(const int* __restrict__ src, const int* __restrict__ dst,
              const float* __restrict__ alpha, const float* __restrict__ alpha_sum,
              const float* __restrict__ h, float* __restrict__ out_h, int Ecnt)
{
    const int t  = threadIdx.x;
    const int ei = blockIdx.x * 4 + (t >> 6);
    if (ei >= Ecnt) return;
    const int c  = (t & 63) * 4;          // 0..252, stays within one head block
    const int hh = c >> 6;
    const int s  = src[ei];
    const int d  = dst[ei];
    const float a  = alpha[(size_t)ei * HEADS + hh];
    const float an = a / (alpha_sum[(size_t)d * HEADS + hh] + 1e-8f);
    const float4 hv = *(const float4*)(h + (size_t)s * COLS + c);
    float* op = out_h + (size_t)d * COLS + c;
    atomicAdd(op + 0, an * hv.x);
    atomicAdd(op + 1, an * hv.y);
    atomicAdd(op + 2, an * hv.z);
    atomicAdd(op + 3, an * hv.w);
}

// ---------------------------------------------------------------------------
extern "C" void kernel_launch(void* const* d_in, const int* in_sizes, int n_in,
                              void* d_out, int out_size, void* d_ws, size_t ws_size,
                              hipStream_t stream)
{
    const float* x        = (const float*)d_in[0];
    const int*   edge_idx = (const int*)d_in[1];
    const float* W        = (const float*)d_in[2];
    const float* attn     = (const float*)d_in[3];
    const float* W_out    = (const float*)d_in[4];
    float*       out      = (float*)d_out;

    const int Nn   = in_sizes[0] / COLS;   // 50000
    const int Ecnt = in_sizes[1] / 2;      // 1000000

    const int* src = edge_idx;
    const int* dst = edge_idx + Ecnt;

    // workspace layout (floats)
    float* ws        = (float*)d_ws;
    float* h         = ws;  ws += (size_t)Nn * COLS;
    float* out_h     = ws;  ws += (size_t)Nn * COLS;
    float* s_src     = ws;  ws += (size_t)Nn * HEADS;
    float* s_dst     = ws;  ws += (size_t)Nn * HEADS;
    float* e_alpha   = ws;  ws += (size_t)Ecnt * HEADS;
    float* e_max     = ws;  ws += (size_t)Nn * HEADS;
    float* alpha_sum = ws;  ws += (size_t)Nn * HEADS;
    float* Bpack     = ws;  ws += (size_t)COLS * COLS;   // reused for both GEMMs

    const int mtiles = (Nn + 15) / 16;
    const dim3 ggrid(mtiles, COLS / 64);

    // 0) init accumulators
    gat_init<<<(Nn * COLS + 255) / 256, 256, 0, stream>>>(out_h, alpha_sum, e_max, Nn);

    // 1) h = x @ W  (W is (H,IN,D): strideI=64, strideH=16384)
    gat_pack_B<<<(COLS * COLS + 255) / 256, 256, 0, stream>>>(W, Bpack, DDIM, 256 * DDIM);
    gat_gemm_wmma_f32<<<ggrid, 32, 0, stream>>>(x, Bpack, h, Nn);

    // 2) per-node scores
    gat_scores<<<(Nn * HEADS + 255) / 256, 256, 0, stream>>>(h, attn, s_src, s_dst, Nn);

    // 3) edge scores + segment max
    gat_edge_score<<<(Ecnt + 255) / 256, 256, 0, stream>>>(
        src, dst, s_src, s_dst, e_alpha, e_max, Ecnt);

    // 4) alpha + segment sum
    gat_edge_alpha<<<(Ecnt + 255) / 256, 256, 0, stream>>>(
        dst, e_alpha, e_max, alpha_sum, Ecnt);

    // 5) normalized scatter-add of messages
    gat_aggregate<<<(Ecnt + 3) / 4, 256, 0, stream>>>(
        src, dst, e_alpha, alpha_sum, h, out_h, Ecnt);

    // 6) out = out_h @ W_out (row-major 256x256: strideI=256, strideH=64)
    gat_pack_B<<<(COLS * COLS + 255) / 256, 256, 0, stream>>>(W_out, Bpack, 256, DDIM);
    gat_gemm_wmma_f32<<<ggrid, 32, 0, stream>>>(out_h, Bpack, out, Nn);
}